// ModelNew_55611236549376
// MI455X (gfx1250) — compile-verified
//
#include <hip/hip_runtime.h>
#include <math.h>

// Segment softmax over CSR rows, 8 heads, fp32. Memory-bound (~205 MB total
// traffic -> ~9 us at 23.3 TB/s). One wave32 per row; lane = (edge_slot, head).
// CDNA5 paths used: wave32 shuffle butterflies, per-wave LDS tile, and the
// gfx1250 async path GLOBAL_LOAD_ASYNC_TO_LDS_B128 + s_wait_asynccnt.

#define H               8
#define TILE            128                 // max edges staged per row in LDS
#define WAVES_PER_BLOCK 8
#define BLOCK           (WAVES_PER_BLOCK * 32)
#define NEG_BIG         (-3.0e38f)          // sentinel (avoids -inf - -inf = NaN)

#if defined(__gfx1250__) &&                                            \
    __has_builtin(__builtin_amdgcn_global_load_async_to_lds_b128) &&   \
    __has_builtin(__builtin_amdgcn_s_wait_asynccnt)
#define USE_ASYNC_LDS 1
#else
#define USE_ASYNC_LDS 0
#endif

#if USE_ASYNC_LDS
typedef int v4i_t __attribute__((ext_vector_type(4)));
typedef __attribute__((address_space(1))) v4i_t* gv4_t;   // global int4*
typedef __attribute__((address_space(3))) v4i_t* lv4_t;   // lds    int4*
#endif

__device__ __forceinline__ void merge_ms(float& m, float& s, float m2, float s2) {
    float mn = fmaxf(m, m2);
    s = s * __expf(m - mn) + s2 * __expf(m2 - mn);
    m = mn;
}

__global__ __launch_bounds__(BLOCK) void seg_softmax_kernel(
    const int* __restrict__ row_ptr,
    const float* __restrict__ scores,
    float* __restrict__ out,
    int n_rows)
{
    __shared__ __align__(16) float lds[WAVES_PER_BLOCK][TILE * H];  // 32 KB/block

    const int wave = threadIdx.x >> 5;
    const int lane = threadIdx.x & 31;
    const int row  = blockIdx.x * WAVES_PER_BLOCK + wave;
    if (row >= n_rows) return;

    const int start = row_ptr[row];
    const int deg   = row_ptr[row + 1] - start;
    if (deg <= 0) return;

    const int h  = lane & 7;    // head
    const int eo = lane >> 3;   // edge sub-slot 0..3  (4 edges x 8 heads / wave iter)

    const float* base  = scores + (size_t)start * H;
    float*       obase = out    + (size_t)start * H;

    float m = NEG_BIG, s = 0.0f;

    if (deg <= TILE) {
        // ---- stage this row's scores into the wave-private LDS slab ----
        float* tile = lds[wave];
        const int nchunk = deg * 2;                 // 16-byte chunks (32 B/edge)
        for (int c = lane; c < nchunk; c += 32) {
#if USE_ASYNC_LDS
            __builtin_amdgcn_global_load_async_to_lds_b128(
                (gv4_t)(void*)(base + c * 4),       // src: global, 16B chunk
                (lv4_t)(void*)(tile + c * 4),       // dst: LDS, 16B aligned
                0, 0);
#else
            const float4 v = *(const float4*)(base + c * 4);
            *(float4*)(tile + c * 4) = v;
#endif
        }
#if USE_ASYNC_LDS
        __builtin_amdgcn_s_wait_asynccnt(0);
#endif
        // same wave wrote & reads its own slab -> no workgroup barrier needed

        // ---- online max/sum (flash-style), 128 B coalesced footprint/iter ----
        for (int e = eo; e < deg; e += 4) {
            float x  = tile[e * H + h];
            float mn = fmaxf(m, x);
            s = s * __expf(m - mn) + __expf(x - mn);
            m = mn;
        }
        // butterfly merge across the 4 edge slots sharing head h (lanes ^8, ^16)
        merge_ms(m, s, __shfl_xor(m, 8, 32),  __shfl_xor(s, 8, 32));
        merge_ms(m, s, __shfl_xor(m, 16, 32), __shfl_xor(s, 16, 32));

        const float inv = 1.0f / s;
        for (int e = eo; e < deg; e += 4) {
            float x = tile[e * H + h];
            obase[e * H + h] = __expf(x - m) * inv;
        }
    } else {
        // ---- rare fallback for huge rows: pure-global two-pass ----
        for (int e = eo; e < deg; e += 4) {
            float x  = base[e * H + h];
            float mn = fmaxf(m, x);
            s = s * __expf(m - mn) + __expf(x - mn);
            m = mn;
        }
        merge_ms(m, s, __shfl_xor(m, 8, 32),  __shfl_xor(s, 8, 32));
        merge_ms(m, s, __shfl_xor(m, 16, 32), __shfl_xor(s, 16, 32));

        const float inv = 1.0f / s;
        for (int e = eo; e < deg; e += 4) {
            float x = base[e * H + h];          // second read: WGP$/L2 hit
            obase[e * H + h] = __expf(x - m) * inv;
        }
    }
}

extern "C" void kernel_launch(void* const* d_in, const int* in_sizes, int n_in,
                              void* d_out, int out_size, void* d_ws, size_t ws_size,
                              hipStream_t stream) {
    const int*   row_ptr = (const int*)d_in[0];     // [N+1] int32
    const float* scores  = (const float*)d_in[1];   // [E, 8] fp32
    float*       out     = (float*)d_out;           // [E, 8] fp32

    const int n_rows = in_sizes[0] - 1;
    const int blocks = (n_rows + WAVES_PER_BLOCK - 1) / WAVES_PER_BLOCK;

    hipLaunchKernelGGL(seg_softmax_kernel, dim3(blocks), dim3(BLOCK), 0, stream,
                       row_ptr, scores, out, n_rows);
}